// SparseTransformer_19971597926621
// MI455X (gfx1250) — compile-verified
//
#include <hip/hip_runtime.h>
#include <math.h>
#include <stdint.h>

#define NVOX  65536
#define CCH   64
#define HH    4
#define DD    16
#define KVOL5 125
#define NPAIR 1048576

typedef __attribute__((ext_vector_type(2))) float v2f;
typedef __attribute__((ext_vector_type(8))) float v8f;

__device__ __forceinline__ v8f wmma_f32(v2f a, v2f b, v8f c) {
    // V_WMMA_F32_16X16X4_F32 : D = A(16x4 f32) x B(4x16 f32) + C(16x16 f32)
    return __builtin_amdgcn_wmma_f32_16x16x4_f32(
        /*neg_a=*/false, a, /*neg_b=*/false, b,
        /*c_mod=*/(short)0, c, /*reuse_a=*/false, /*reuse_b=*/false);
}

// fp32 scatter-add. Differential-verified (round 3 vs 4) to lower to a single
// hardware GLOBAL_ATOMIC_ADD_F32 (no-return, STOREcnt-tracked) with the
// per-element displacement folded into the instruction immediate offset.
__device__ __forceinline__ void atomic_add_f32_dev(float* p, float val) {
    __hip_atomic_fetch_add(p, val, __ATOMIC_RELAXED, __HIP_MEMORY_SCOPE_AGENT);
}

// LDS weight layout: K-pair packed so a B-fragment (B[k][c], B[k+1][c]) is one
// aligned 8-byte ds_load_b64:  lds[(k>>1)*128 + c*2 + (k&1)] = W[k*64 + c]
__device__ __forceinline__ int wpack(int i) {
    const int kk = i >> 6;
    const int c  = i & 63;
    return ((kk >> 1) << 7) + (c << 1) + (kk & 1);
}

// ---------------------------------------------------------------------------
// Pair phase: attn[p,h] = <nq[out,h], nk[in,h] + npos[kidx,h]>
//             accum[out,h,:] += attn * v[in,h,:]   (fp32 HW atomics)
// One thread per (pair, head). 4 consecutive threads share `out` row, so the
// 16 atomic adds per thread form a contiguous 256B burst per pair.
// ---------------------------------------------------------------------------
__global__ __launch_bounds__(256)
void pair_kernel(const int* __restrict__ kq,
                 const float* __restrict__ nq, const float* __restrict__ nk,
                 const float* __restrict__ npos, const float* __restrict__ v,
                 float* __restrict__ acc)
{
    const int gid = blockIdx.x * 256 + threadIdx.x;
    const int p = gid >> 2;
    const int h = gid & 3;

    const int kq0 = kq[p];
    const int out = kq[NPAIR + p];
    const int in  = kq0 / KVOL5;
    const int kid = kq0 - in * KVOL5;

    const float4* __restrict__ qp = (const float4*)(nq   + (size_t)out * CCH + h * DD);
    const float4* __restrict__ kp = (const float4*)(nk   + (size_t)in  * CCH + h * DD);
    const float4* __restrict__ pp = (const float4*)(npos + (size_t)kid * CCH + h * DD);
    const float4* __restrict__ vp = (const float4*)(v    + (size_t)in  * CCH + h * DD);

    float att = 0.0f;
    float4 vv[4];
#pragma unroll
    for (int i = 0; i < 4; ++i) {
        const float4 a = qp[i];
        const float4 b = kp[i];
        const float4 c = pp[i];
        att += a.x * (b.x + c.x) + a.y * (b.y + c.y)
             + a.z * (b.z + c.z) + a.w * (b.w + c.w);
        vv[i] = vp[i];
    }

    float* __restrict__ d = acc + (size_t)out * CCH + h * DD;
#pragma unroll
    for (int i = 0; i < 4; ++i) {
        atomic_add_f32_dev(d + 4 * i + 0, att * vv[i].x);
        atomic_add_f32_dev(d + 4 * i + 1, att * vv[i].y);
        atomic_add_f32_dev(d + 4 * i + 2, att * vv[i].z);
        atomic_add_f32_dev(d + 4 * i + 3, att * vv[i].w);
    }
}

// ---------------------------------------------------------------------------
// Fused Q/K/V GEMM: q|k|v = x @ W{q,k,v} + b{q,k,v}
// One wave per 16-row tile; weights (3 x 16KB) staged K-pair-packed in LDS.
// A-fragment (16x4 f32): lane 0-15 -> M=lane, K={0,1}; lane 16-31 -> K={2,3}
// B-fragment (4x16 f32): lane 0-15 -> N=lane, K={0,1}; lane 16-31 -> K={2,3}
// C/D (16x16 f32): vgpr r, lanes0-15 -> (M=r, N=lane); lanes16-31 -> (M=r+8)
// ---------------------------------------------------------------------------
__global__ __launch_bounds__(256)
void qkv_kernel(const float* __restrict__ x,
                const float* __restrict__ Wq, const float* __restrict__ bq,
                const float* __restrict__ Wk, const float* __restrict__ bk,
                const float* __restrict__ Wv, const float* __restrict__ bv,
                float* __restrict__ q, float* __restrict__ k, float* __restrict__ v)
{
    __shared__ float w[3 * CCH * CCH];   // 48 KB, K-pair packed
    const int tid = threadIdx.x;
    for (int i = tid; i < CCH * CCH; i += 256) {
        const int d = wpack(i);
        w[d]                 = Wq[i];
        w[CCH * CCH + d]     = Wk[i];
        w[2 * CCH * CCH + d] = Wv[i];
    }
    __syncthreads();

    const int wave = tid >> 5;
    const int lane = tid & 31;
    const int m0   = (blockIdx.x * 8 + wave) * 16;
    const int lrow = lane & 15;
    const int klo  = (lane >> 4) << 1;      // 0 for lanes 0-15, 2 for 16-31

    v8f aq[4] = {}, ak[4] = {}, av[4] = {};

    const float* __restrict__ arow = x + (size_t)(m0 + lrow) * CCH;

#pragma unroll 4
    for (int kk = 0; kk < CCH; kk += 4) {
        v2f a;
        a.x = arow[kk + klo];
        a.y = arow[kk + klo + 1];
        const int kp = ((kk + klo) >> 1) << 7;     // packed pair row base
#pragma unroll
        for (int j = 0; j < 4; ++j) {
            const int c2 = ((j << 4) + lrow) << 1;
            v2f b;
            b = *(const v2f*)&w[kp + c2];
            aq[j] = wmma_f32(a, b, aq[j]);
            b = *(const v2f*)&w[CCH * CCH + kp + c2];
            ak[j] = wmma_f32(a, b, ak[j]);
            b = *(const v2f*)&w[2 * CCH * CCH + kp + c2];
            av[j] = wmma_f32(a, b, av[j]);
        }
    }

    const int rowoff = (lane >> 4) * 8;      // 0 or 8
#pragma unroll
    for (int j = 0; j < 4; ++j) {
        const int c = j * 16 + lrow;
        const float bqv = bq[c], bkv = bk[c], bvv = bv[c];
#pragma unroll
        for (int r = 0; r < 8; ++r) {
            const size_t o = (size_t)(m0 + rowoff + r) * CCH + c;
            q[o] = aq[j][r] + bqv;
            k[o] = ak[j][r] + bkv;
            v[o] = av[j][r] + bvv;
        }
    }
}

// ---------------------------------------------------------------------------
// Row-wise L2 normalize (dim = 16). q,k normalized in place; pos_enc -> npos.
// ---------------------------------------------------------------------------
__global__ __launch_bounds__(256)
void normalize_kernel(float* __restrict__ q, float* __restrict__ k,
                      const float* __restrict__ pos, float* __restrict__ npos)
{
    const int gid = blockIdx.x * 256 + threadIdx.x;
    const int nq_rows = NVOX * HH;
    const int total = 2 * nq_rows + KVOL5 * HH;
    if (gid >= total) return;

    const float* src;
    float* dst;
    if (gid < nq_rows)            { src = q + (size_t)gid * DD;             dst = q + (size_t)gid * DD; }
    else if (gid < 2 * nq_rows)   { const int r = gid - nq_rows;
                                    src = k + (size_t)r * DD;               dst = k + (size_t)r * DD; }
    else                          { const int r = gid - 2 * nq_rows;
                                    src = pos + (size_t)r * DD;             dst = npos + (size_t)r * DD; }

    float4 a0 = ((const float4*)src)[0];
    float4 a1 = ((const float4*)src)[1];
    float4 a2 = ((const float4*)src)[2];
    float4 a3 = ((const float4*)src)[3];
    float s = a0.x*a0.x + a0.y*a0.y + a0.z*a0.z + a0.w*a0.w
            + a1.x*a1.x + a1.y*a1.y + a1.z*a1.z + a1.w*a1.w
            + a2.x*a2.x + a2.y*a2.y + a2.z*a2.z + a2.w*a2.w
            + a3.x*a3.x + a3.y*a3.y + a3.z*a3.z + a3.w*a3.w;
    const float inv = 1.0f / fmaxf(sqrtf(s), 1e-12f);
    a0.x *= inv; a0.y *= inv; a0.z *= inv; a0.w *= inv;
    a1.x *= inv; a1.y *= inv; a1.z *= inv; a1.w *= inv;
    a2.x *= inv; a2.y *= inv; a2.z *= inv; a2.w *= inv;
    a3.x *= inv; a3.y *= inv; a3.z *= inv; a3.w *= inv;
    ((float4*)dst)[0] = a0; ((float4*)dst)[1] = a1;
    ((float4*)dst)[2] = a2; ((float4*)dst)[3] = a3;
}

// ---------------------------------------------------------------------------
// Output projection: out = accum @ Wo + bo + x   (fused bias + residual)
// ---------------------------------------------------------------------------
__global__ __launch_bounds__(256)
void proj_kernel(const float* __restrict__ accin, const float* __restrict__ Wo,
                 const float* __restrict__ bo, const float* __restrict__ x,
                 float* __restrict__ out)
{
    __shared__ float w[CCH * CCH];       // 16 KB, K-pair packed
    const int tid = threadIdx.x;
    for (int i = tid; i < CCH * CCH; i += 256) w[wpack(i)] = Wo[i];
    __syncthreads();

    const int wave = tid >> 5;
    const int lane = tid & 31;
    const int m0   = (blockIdx.x * 8 + wave) * 16;
    const int lrow = lane & 15;
    const int klo  = (lane >> 4) << 1;

    v8f accd[4] = {};
    const float* __restrict__ arow = accin + (size_t)(m0 + lrow) * CCH;

#pragma unroll 4
    for (int kk = 0; kk < CCH; kk += 4) {
        v2f a;
        a.x = arow[kk + klo];
        a.y = arow[kk + klo + 1];
        const int kp = ((kk + klo) >> 1) << 7;
#pragma unroll
        for (int j = 0; j < 4; ++j) {
            const int c2 = ((j << 4) + lrow) << 1;
            const v2f b = *(const v2f*)&w[kp + c2];
            accd[j] = wmma_f32(a, b, accd[j]);
        }
    }

    const int rowoff = (lane >> 4) * 8;
#pragma unroll
    for (int j = 0; j < 4; ++j) {
        const int c = j * 16 + lrow;
        const float bov = bo[c];
#pragma unroll
        for (int r = 0; r < 8; ++r) {
            const size_t o = (size_t)(m0 + rowoff + r) * CCH + c;
            out[o] = accd[j][r] + bov + x[o];
        }
    }
}

// ---------------------------------------------------------------------------
extern "C" void kernel_launch(void* const* d_in, const int* in_sizes, int n_in,
                              void* d_out, int out_size, void* d_ws, size_t ws_size,
                              hipStream_t stream)
{
    const float* x   = (const float*)d_in[0];
    const float* Wq  = (const float*)d_in[1];
    const float* bq  = (const float*)d_in[2];
    const float* Wk  = (const float*)d_in[3];
    const float* bk  = (const float*)d_in[4];
    const float* Wv  = (const float*)d_in[5];
    const float* bv  = (const float*)d_in[6];
    const float* Wo  = (const float*)d_in[7];
    const float* bo  = (const float*)d_in[8];
    const float* pos = (const float*)d_in[9];
    const int*   kq  = (const int*)d_in[10];
    float* out = (float*)d_out;

    // Workspace layout (floats): q | k | v | accum | npos  (~64 MB)
    float* qws    = (float*)d_ws;
    float* kws    = qws  + (size_t)NVOX * CCH;
    float* vws    = kws  + (size_t)NVOX * CCH;
    float* accws  = vws  + (size_t)NVOX * CCH;
    float* nposws = accws + (size_t)NVOX * CCH;

    hipMemsetAsync(accws, 0, (size_t)NVOX * CCH * sizeof(float), stream);

    qkv_kernel<<<NVOX / 128, 256, 0, stream>>>(x, Wq, bq, Wk, bk, Wv, bv,
                                               qws, kws, vws);

    const int rows = 2 * NVOX * HH + KVOL5 * HH;
    normalize_kernel<<<(rows + 255) / 256, 256, 0, stream>>>(qws, kws, pos, nposws);

    pair_kernel<<<(NPAIR * HH) / 256, 256, 0, stream>>>(kq, qws, kws, nposws,
                                                        vws, accws);

    proj_kernel<<<NVOX / 128, 256, 0, stream>>>(accws, Wo, bo, x, out);
}